// MambaStack_76768245449520
// MI455X (gfx1250) — compile-verified
//
#include <hip/hip_runtime.h>
#include <hip/hip_bf16.h>
#include <math.h>

// ---------------------------------------------------------------------------
// MambaStack forward for MI455X (gfx1250, wave32).
// GEMMs: V_WMMA_F32_16X16X4_F32 fed from a double-buffered LDS pipeline that
// is filled with GLOBAL_LOAD_ASYNC_TO_LDS_B128 (ASYNCcnt) so tile DMA overlaps
// the matrix pipe. Selective scan is sequential over L with LDS broadcast of
// the shared B/C state vectors.
// ---------------------------------------------------------------------------

#define D_MODEL  512
#define D_INNER  1024
#define D_STATE  16
#define DT_RANK  32
#define D_CONV   4
#define N_LAYER  4
#define SEQ_L    1024
#define BATCH    2
#define M_TOK    (BATCH * SEQ_L)   // 2048 token rows

#define BK   32                    // K-tile depth
#define PAD  36                    // padded LDS row stride in floats:
                                   //  - 144 B rows keep b128 stores 16B-aligned
                                   //  - 36*r mod 64 hits 16 distinct banks for r=0..15

typedef __attribute__((ext_vector_type(2)))  float  v2f;
typedef __attribute__((ext_vector_type(8)))  float  v8f;
typedef int b128_t __attribute__((vector_size(16)));   // matches builtin param type

#if defined(__has_builtin)
#  if __has_builtin(__builtin_amdgcn_global_load_async_to_lds_b128)
#    define HAS_ASYNC_LDS 1
#  else
#    define HAS_ASYNC_LDS 0
#  endif
#else
#  define HAS_ASYNC_LDS 0
#endif

#if HAS_ASYNC_LDS
#  if __has_builtin(__builtin_amdgcn_s_wait_asynccnt)
#    define WAIT_ASYNC(n) __builtin_amdgcn_s_wait_asynccnt(n)
#  else
#    define WAIT_ASYNC(n) asm volatile("s_wait_asynccnt %0" :: "n"(n))
#  endif
#else
#  define WAIT_ASYNC(n) ((void)0)
#endif

// One 16-byte global -> LDS transfer per lane (ASYNCcnt-tracked DMA).
__device__ inline void cp_b128(const float* g, float* l) {
#if HAS_ASYNC_LDS
  __builtin_amdgcn_global_load_async_to_lds_b128(
      (__attribute__((address_space(1))) b128_t*)g,
      (__attribute__((address_space(3))) b128_t*)l,
      0, 0);
#else
  *(float4*)l = *(const float4*)g;   // synchronous fallback
#endif
}

__device__ inline float softplusf(float x) {
  return (x > 20.f) ? x : log1pf(expf(x));
}
__device__ inline float siluf(float x) {
  return x / (1.f + expf(-x));
}

// Block-wide (256 threads = 8 waves) all-reduce sum, wave32 shuffles + LDS.
__device__ inline float block_allreduce(float v) {
  __shared__ float red[8];
  #pragma unroll
  for (int off = 16; off > 0; off >>= 1) v += __shfl_xor(v, off, 32);
  if ((threadIdx.x & 31) == 0) red[threadIdx.x >> 5] = v;
  __syncthreads();
  if (threadIdx.x == 0) {
    float s = 0.f;
    #pragma unroll
    for (int i = 0; i < 8; i++) s += red[i];
    red[0] = s;
  }
  __syncthreads();
  float s = red[0];
  __syncthreads();   // make red[] safe for reuse by a later call
  return s;
}

// ---------------------------------------------------------------------------
// Embed: hs[m][d] = sum_c ids[m][c] * W_in[d][c]   (K=3, trivial)
// ---------------------------------------------------------------------------
__global__ void k_embed(const float* __restrict__ ids,
                        const float* __restrict__ W_in,
                        float* __restrict__ hs) {
  int t = blockIdx.x * blockDim.x + threadIdx.x;   // M_TOK * D_MODEL
  int m = t / D_MODEL;
  int d = t - m * D_MODEL;
  const float* ir = ids + (size_t)m * 3;
  const float* wr = W_in + (size_t)d * 3;
  hs[t] = ir[0] * wr[0] + ir[1] * wr[1] + ir[2] * wr[2];
}

// ---------------------------------------------------------------------------
// Fused residual add + RMSNorm (D = 512, block = 256 threads, 1 row / block)
// ---------------------------------------------------------------------------
__global__ __launch_bounds__(256)
void k_addnorm(const float* __restrict__ hs,
               const float* __restrict__ resid_in,
               float* __restrict__ resid_out,
               float* __restrict__ out_norm,
               const float* __restrict__ w) {
  const int row = blockIdx.x;
  const int tid = threadIdx.x;
  const size_t base = (size_t)row * D_MODEL;
  float v0 = hs[base + tid];
  float v1 = hs[base + tid + 256];
  if (resid_in) {
    v0 += resid_in[base + tid];
    v1 += resid_in[base + tid + 256];
  }
  float ss = block_allreduce(v0 * v0 + v1 * v1);
  float rms = rsqrtf(ss * (1.f / D_MODEL) + 1e-5f);
  resid_out[base + tid]       = v0;
  resid_out[base + tid + 256] = v1;
  out_norm[base + tid]        = v0 * rms * w[tid];
  out_norm[base + tid + 256]  = v1 * rms * w[tid + 256];
}

// ---------------------------------------------------------------------------
// WMMA GEMM:  C[M,N] = A[M,K] * W[N,K]^T     (all fp32)
//
// Block = 256 threads = 8 waves arranged 2(M) x 4(N); block tile 32 x 128,
// each wave computes a 16x32 output tile (two v8f accumulators).
// K is processed in BK=32 tiles through a double-buffered LDS stage filled by
// GLOBAL_LOAD_ASYNC_TO_LDS_B128 (5 async ops per wave per tile), overlapping
// the DMA of tile t+1 with the 16 V_WMMA_F32_16X16X4_F32 of tile t.
// Requires M%32==0, N%32==0, K%32==0 (true for every projection here).
// Out-of-range B rows (N=64 case) are clamped at the staging loads (keeps the
// per-wave ASYNCcnt uniform at 5) and only the store phase is predicated, so
// every wave participates in all barriers and EXEC is all-ones at the WMMAs.
// epi==1: C = softplus(C + bias[n])   (fused dt-projection epilogue)
// ---------------------------------------------------------------------------
__global__ __launch_bounds__(256)
void k_gemm(const float* __restrict__ A, int lda,
            const float* __restrict__ W, int ldw,
            float* __restrict__ C, int ldc,
            int N, int K,
            const float* __restrict__ bias, int epi) {
  __shared__ float As[2][32 * PAD];    //  4.5 KB per buffer
  __shared__ float Bs[2][128 * PAD];   // 18   KB per buffer  (total 45 KB)

  const int tid  = threadIdx.x;
  const int lane = tid & 31;
  const int wave = tid >> 5;
  const int wm = wave & 1;
  const int wn = wave >> 1;
  const int mBlk = blockIdx.y * 32;
  const int nBlk = blockIdx.x * 128;
  const int r = lane & 15;             // row-within-fragment index
  const int h = lane >> 4;             // lane-half selects K offset

  // ---- staging: 256 threads cover A(32x32) with 1 b128 each and B(128x32)
  //      with 4 b128 each; per-wave async-op count is exactly 5.
  const int srow   = tid >> 3;         // 0..31
  const int schunk = (tid & 7) * 4;    // float column 0,4,...,28
  const float* gA0 = A + (size_t)(mBlk + srow) * lda + schunk;
  float* lA0 = &As[0][srow * PAD + schunk];
  float* lB0 = &Bs[0][srow * PAD + schunk];

  auto stageTile = [&](int kt, int buf) {
    const int kofs = kt * BK;
    cp_b128(gA0 + kofs, lA0 + buf * (32 * PAD));
    #pragma unroll
    for (int i = 0; i < 4; i++) {
      const int brow = srow + i * 32;
      const int gn   = nBlk + brow;
      const int gnc  = (gn < N) ? gn : 0;      // clamp: always issue, stay safe
      cp_b128(W + (size_t)gnc * ldw + kofs + schunk,
              lB0 + buf * (128 * PAD) + i * (32 * PAD));
    }
  };

  const int nTiles = K / BK;
  v8f acc0 = {};
  v8f acc1 = {};

  stageTile(0, 0);
  for (int t = 0; t < nTiles; t++) {
    const int buf = t & 1;
    if (t + 1 < nTiles) {
      stageTile(t + 1, buf ^ 1);
      WAIT_ASYNC(5);                   // tile t's 5 ops done; t+1 in flight
    } else {
      WAIT_ASYNC(0);
    }
    __syncthreads();                   // tile t visible to every wave

    const float* pA  = &As[buf][(wm * 16 + r) * PAD];
    const float* pB0 = &Bs[buf][(wn * 32 + r) * PAD];
    const float* pB1 = &Bs[buf][(wn * 32 + 16 + r) * PAD];
    #pragma unroll
    for (int kk = 0; kk < BK; kk += 4) {
      const int ka = kk + 2 * h;
      // A: lane r (M=r), VGPR v -> K = ka + v.  B: VGPR v -> K row, lane -> N.
      v2f a  = *(const v2f*)(pA + ka);
      v2f b0 = *(const v2f*)(pB0 + ka);
      v2f b1 = *(const v2f*)(pB1 + ka);
      acc0 = __builtin_amdgcn_wmma_f32_16x16x4_f32(false, a, false, b0,
                                                   (short)0, acc0, false, false);
      acc1 = __builtin_amdgcn_wmma_f32_16x16x4_f32(false, a, false, b1,
                                                   (short)0, acc1, false, false);
    }
    __syncthreads();                   // reads done before buf is re-filled
  }

  const int n0 = nBlk + wn * 32;
  if (n0 >= N) return;                 // garbage tiles: computed, never stored

  // C/D layout: VGPR i, lanes 0-15 -> M=i, N=lane; lanes 16-31 -> M=i+8.
  const int c0 = n0 + r;
  const int c1 = n0 + 16 + r;
  #pragma unroll
  for (int i = 0; i < 8; i++) {
    const int row = mBlk + wm * 16 + i + 8 * h;
    float o0 = acc0[i];
    float o1 = acc1[i];
    if (epi == 1) {
      o0 = softplusf(o0 + bias[c0]);
      o1 = softplusf(o1 + bias[c1]);
    }
    C[(size_t)row * ldc + c0] = o0;
    C[(size_t)row * ldc + c1] = o1;
  }
}

// ---------------------------------------------------------------------------
// Depthwise causal conv (kernel 4) + SiLU.
// x lives in xz[..., :D_INNER] (row stride 2*D_INNER); output -> xc (dense).
// ---------------------------------------------------------------------------
__global__ void k_conv_silu(const float* __restrict__ xz,
                            const float* __restrict__ cw,
                            const float* __restrict__ cb,
                            float* __restrict__ xc) {
  int t = blockIdx.x * blockDim.x + threadIdx.x;   // M_TOK * D_INNER
  int d = t & (D_INNER - 1);
  int m = t >> 10;
  int l = m & (SEQ_L - 1);
  const float* base = xz + (size_t)m * (2 * D_INNER) + d;
  float acc = cb[d];
  #pragma unroll
  for (int j = 0; j < D_CONV; j++) {
    int tp = l - (D_CONV - 1) + j;
    float xv = (tp >= 0) ? base[(ptrdiff_t)(j - (D_CONV - 1)) * (2 * D_INNER)]
                         : 0.f;
    acc += cw[d * D_CONV + j] * xv;
  }
  xc[t] = siluf(acc);
}

// ---------------------------------------------------------------------------
// Selective scan, fused with SiLU(z) gating.
// Block = 256 threads (256 channels of one batch element); grid = 8.
// Per timestep the shared (b,l) B/C vectors are staged once in LDS and
// broadcast to all channels; the 16-state recurrence stays in VGPRs.
// ---------------------------------------------------------------------------
__global__ __launch_bounds__(256)
void k_scan(const float* __restrict__ xc,
            const float* __restrict__ dtb,
            const float* __restrict__ xz,
            const float* __restrict__ xdbl,
            const float* __restrict__ A_log,
            const float* __restrict__ Dskip,
            float* __restrict__ y) {
  const int b = blockIdx.x >> 2;
  const int d = ((blockIdx.x & 3) << 8) + threadIdx.x;

  float Ad[D_STATE];
  #pragma unroll
  for (int n = 0; n < D_STATE; n++) Ad[n] = -expf(A_log[d * D_STATE + n]);
  const float Dd = Dskip[d];

  float hst[D_STATE];
  #pragma unroll
  for (int n = 0; n < D_STATE; n++) hst[n] = 0.f;

  __shared__ float sB[D_STATE];
  __shared__ float sC[D_STATE];

  for (int l = 0; l < SEQ_L; l++) {
    const int m = (b << 10) + l;
    if (threadIdx.x < 2 * D_STATE) {
      float v = xdbl[(size_t)m * 64 + DT_RANK + threadIdx.x];
      if (threadIdx.x < D_STATE) sB[threadIdx.x] = v;
      else                       sC[threadIdx.x - D_STATE] = v;
    }
    __syncthreads();

    const float dtv = dtb[(size_t)m * D_INNER + d];
    const float xv  = xc [(size_t)m * D_INNER + d];
    const float zv  = xz [(size_t)m * (2 * D_INNER) + D_INNER + d];

    float acc = 0.f;
    #pragma unroll
    for (int n = 0; n < D_STATE; n++) {
      float dA = expf(dtv * Ad[n]);
      hst[n] = dA * hst[n] + (dtv * sB[n]) * xv;
      acc += hst[n] * sC[n];
    }
    float yv = (acc + Dd * xv) * siluf(zv);
    y[(size_t)m * D_INNER + d] = yv;
    __syncthreads();   // protect sB/sC before next timestep's staging
  }
}

// ---------------------------------------------------------------------------
// Head: out[m][j] = sum_d hsn[m][d] * W_out[j][d]   (j < 3)
// ---------------------------------------------------------------------------
__global__ __launch_bounds__(256)
void k_head(const float* __restrict__ hsn,
            const float* __restrict__ Wout,
            float* __restrict__ out) {
  const int row = blockIdx.x;
  const int tid = threadIdx.x;
  const size_t base = (size_t)row * D_MODEL;
  float v0 = hsn[base + tid];
  float v1 = hsn[base + tid + 256];
  float s0 = v0 * Wout[tid]             + v1 * Wout[tid + 256];
  float s1 = v0 * Wout[D_MODEL + tid]   + v1 * Wout[D_MODEL + tid + 256];
  float s2 = v0 * Wout[2*D_MODEL + tid] + v1 * Wout[2*D_MODEL + tid + 256];
  s0 = block_allreduce(s0);
  s1 = block_allreduce(s1);
  s2 = block_allreduce(s2);
  if (tid == 0) {
    out[row * 3 + 0] = s0;
    out[row * 3 + 1] = s1;
    out[row * 3 + 2] = s2;
  }
}

// ---------------------------------------------------------------------------
extern "C" void kernel_launch(void* const* d_in, const int* in_sizes, int n_in,
                              void* d_out, int out_size, void* d_ws, size_t ws_size,
                              hipStream_t stream) {
  (void)in_sizes; (void)n_in; (void)out_size; (void)ws_size;

  const float* input_ids = (const float*)d_in[0];
  const float* W_in      = (const float*)d_in[1];
  const float* W_out     = (const float*)d_in[2];
  const float* norm_w    = (const float*)d_in[3];
  const float* in_proj_W = (const float*)d_in[4];
  const float* conv_w    = (const float*)d_in[5];
  const float* conv_b    = (const float*)d_in[6];
  const float* x_proj_W  = (const float*)d_in[7];
  const float* dt_proj_W = (const float*)d_in[8];
  const float* dt_proj_b = (const float*)d_in[9];
  const float* A_log     = (const float*)d_in[10];
  const float* D_skip    = (const float*)d_in[11];
  const float* out_proj_W= (const float*)d_in[12];
  const float* norm_f_w  = (const float*)d_in[13];

  // Workspace carve-out (floats): ~13.1M floats (~52 MB)
  float* ws    = (float*)d_ws;
  float* resid = ws;                               // M_TOK * D_MODEL
  float* hsn   = resid + (size_t)M_TOK * D_MODEL;  // M_TOK * D_MODEL
  float* hs    = hsn   + (size_t)M_TOK * D_MODEL;  // M_TOK * D_MODEL
  float* xz    = hs    + (size_t)M_TOK * D_MODEL;  // M_TOK * 2*D_INNER
  float* xc    = xz    + (size_t)M_TOK * 2 * D_INNER;
  float* xdbl  = xc    + (size_t)M_TOK * D_INNER;  // M_TOK * 64
  float* dtb   = xdbl  + (size_t)M_TOK * 64;       // M_TOK * D_INNER
  float* yb    = dtb   + (size_t)M_TOK * D_INNER;  // M_TOK * D_INNER

  // Embed: (B,L,3) @ W_in^T -> hs (M_TOK, D_MODEL)
  k_embed<<<(M_TOK * D_MODEL) / 256, 256, 0, stream>>>(input_ids, W_in, hs);

  const dim3 gIn (2 * D_INNER / 128, M_TOK / 32);  // in_proj:  N=2048
  const dim3 gXp (1,                 M_TOK / 32);  // x_proj:   N=64
  const dim3 gDt (D_INNER / 128,     M_TOK / 32);  // dt_proj:  N=1024
  const dim3 gOut(D_MODEL / 128,     M_TOK / 32);  // out_proj: N=512

  for (int i = 0; i < N_LAYER; i++) {
    // residual add + RMSNorm
    k_addnorm<<<M_TOK, 256, 0, stream>>>(hs, (i == 0) ? nullptr : resid,
                                         resid, hsn, norm_w + (size_t)i * D_MODEL);
    // in_proj: xz = hsn @ in_proj_W^T
    k_gemm<<<gIn, 256, 0, stream>>>(hsn, D_MODEL,
                                    in_proj_W + (size_t)i * 2 * D_INNER * D_MODEL, D_MODEL,
                                    xz, 2 * D_INNER,
                                    2 * D_INNER, D_MODEL, nullptr, 0);
    // causal depthwise conv + SiLU
    k_conv_silu<<<(M_TOK * D_INNER) / 256, 256, 0, stream>>>(
        xz, conv_w + (size_t)i * D_INNER * D_CONV,
        conv_b + (size_t)i * D_INNER, xc);
    // x_proj: xdbl = xc @ x_proj_W^T   (N=64: dt_raw | B | C)
    k_gemm<<<gXp, 256, 0, stream>>>(xc, D_INNER,
                                    x_proj_W + (size_t)i * 64 * D_INNER, D_INNER,
                                    xdbl, 64,
                                    64, D_INNER, nullptr, 0);
    // dt_proj + softplus(.+b): dtb = softplus(xdbl[:, :32] @ dt_proj_W^T + b)
    k_gemm<<<gDt, 256, 0, stream>>>(xdbl, 64,
                                    dt_proj_W + (size_t)i * D_INNER * DT_RANK, DT_RANK,
                                    dtb, D_INNER,
                                    D_INNER, DT_RANK,
                                    dt_proj_b + (size_t)i * D_INNER, 1);
    // selective scan fused with SiLU(z) gating -> yb
    k_scan<<<8, 256, 0, stream>>>(xc, dtb, xz, xdbl,
                                  A_log + (size_t)i * D_INNER * D_STATE,
                                  D_skip + (size_t)i * D_INNER, yb);
    // out_proj: hs = yb @ out_proj_W^T
    k_gemm<<<gOut, 256, 0, stream>>>(yb, D_INNER,
                                     out_proj_W + (size_t)i * D_MODEL * D_INNER, D_INNER,
                                     hs, D_MODEL,
                                     D_MODEL, D_INNER, nullptr, 0);
  }

  // final residual + RMSNorm + head projection
  k_addnorm<<<M_TOK, 256, 0, stream>>>(hs, resid, resid, hsn, norm_f_w);
  k_head<<<M_TOK, 256, 0, stream>>>(hsn, W_out, (float*)d_out);
}